// BACKFLOW_28552942584445
// MI455X (gfx1250) — compile-verified
//
#include <hip/hip_runtime.h>
#include <hip/hip_bf16.h>

#define NUM_ORBITAL 128
#define NUM_ELECTRON 32
#define HID 4
#define LUP 33  // padded row stride: coprime with 64 banks -> conflict-free columns

typedef __attribute__((ext_vector_type(2))) float v2f;
typedef __attribute__((ext_vector_type(8))) float v8f;

// ---------------------------------------------------------------------------
// Kernel 1: jnp.nonzero(x==1, size=32)[1]: column indices of the first 32
// nonzeros in row-major flat order, zero-padded. Data guarantees termination
// after ~32 elements (row 0 cols 0..31 are forced to 1).
// ---------------------------------------------------------------------------
__global__ void bf_find_cols(const float* __restrict__ x, int* __restrict__ cols,
                             long total) {
  if (threadIdx.x == 0 && blockIdx.x == 0) {
    int cnt = 0;
    for (long i = 0; i < total && cnt < NUM_ELECTRON; ++i) {
      if (x[i] == 1.0f) cols[cnt++] = (int)(i % NUM_ORBITAL);
    }
    for (; cnt < NUM_ELECTRON; ++cnt) cols[cnt] = 0;
  }
}

// ---------------------------------------------------------------------------
// Kernel 2: fused per-16-row tile.
//   wave 0: stages zero-padded W1 B-fragments in LDS (exact WMMA B layout),
//           then h1 = tanh(x_tile @ W1 + b1) via 32 accumulating
//           V_WMMA_F32_16X16X4_F32, then h2 = tanh(h1 @ W2 + b2) -> LDS.
//   all 8 waves: 2 batch rows each; build the 32x32 slater matrix (gathering
//           only cols[] orbitals of W3/b3) in a private padded LDS slab,
//           LU with partial pivoting (shfl-reduction argmax), det = Π pivots.
// ---------------------------------------------------------------------------
__launch_bounds__(256)
__global__ void bf_main(const float* __restrict__ x,
                        const float* __restrict__ W1,
                        const float* __restrict__ b1,
                        const float* __restrict__ W2,
                        const float* __restrict__ b2,
                        const float* __restrict__ W3,
                        const float* __restrict__ b3,
                        const int* __restrict__ colsg,
                        float* __restrict__ out) {
  __shared__ float sW1B[(NUM_ORBITAL / 4) * 64];   // 8 KB: per-step B fragments
  __shared__ float sH1[16 * HID];
  __shared__ float sH2[16 * HID];
  __shared__ int   sCols[NUM_ELECTRON];
  __shared__ float sLU[8 * NUM_ELECTRON * LUP];    // ~33 KB: 1 matrix per wave

  const int lane = threadIdx.x & 31;
  const int wid  = threadIdx.x >> 5;   // 0..7
  const int base = blockIdx.x * 16;

  if (threadIdx.x < NUM_ELECTRON) sCols[threadIdx.x] = colsg[threadIdx.x];

  if (wid == 0) {
    // ---- Stage B fragments: sW1B[kk*64 + lane*2 + j] = W1pad[kk*4+koff+j][n]
    //      (n = lane&15, koff = lane<16 ? 0 : 2; zero for n >= HID). ----
    for (int t = lane; t < (NUM_ORBITAL / 4) * 64; t += 32) sW1B[t] = 0.0f;
    for (int e = lane; e < NUM_ORBITAL * HID; e += 32) {  // e = k*HID + n
      const int k = e >> 2, nn = e & 3;
      const int dst = (k >> 2) * 64 + (nn + ((k & 2) ? 16 : 0)) * 2 + (k & 1);
      sW1B[dst] = W1[e];
    }

    // ---- WMMA f32 16x16x4 (ISA 7.12.2 32-bit layouts) ----
    const int n    = (lane < 16) ? lane : (lane - 16);
    const int koff = (lane < 16) ? 0 : 2;
    const float* xrow = x + (size_t)(base + n) * NUM_ORBITAL;

    const float cin = (n < HID) ? b1[n] : 0.0f;
    v8f c = {cin, cin, cin, cin, cin, cin, cin, cin};

    for (int kk = 0; kk < NUM_ORBITAL / 4; ++kk) {
      const v2f a = *(const v2f*)(xrow + kk * 4 + koff);   // 8B aligned
      const v2f b = *(const v2f*)&sW1B[kk * 64 + lane * 2];
      c = __builtin_amdgcn_wmma_f32_16x16x4_f32(false, a, false, b,
                                                (short)0, c, false, false);
    }
#pragma unroll
    for (int r = 0; r < 8; ++r) {
      const int row = (lane < 16) ? r : (r + 8);
      if (n < HID) sH1[row * HID + n] = tanhf(c[r]);
    }
    // h2 = tanh(h1 @ W2 + b2): 16x4 outputs, 32 lanes -> two halves
#pragma unroll
    for (int half = 0; half < 2; ++half) {
      const int m = half * 8 + (lane >> 2);
      const int j = lane & 3;
      float acc = b2[j];
#pragma unroll
      for (int f = 0; f < HID; ++f) acc += sH1[m * HID + f] * W2[f * HID + j];
      sH2[m * HID + j] = tanhf(acc);
    }
  }
  __syncthreads();

  float* buf = &sLU[wid * (NUM_ELECTRON * LUP)];
  for (int rr = 0; rr < 2; ++rr) {
    const int m = wid * 2 + rr;
    const float h0 = sH2[m * HID + 0], h1v = sH2[m * HID + 1];
    const float h2v = sH2[m * HID + 2], h3v = sH2[m * HID + 3];

    // slater[i][lane] = tanh(b3[cols[i],lane] + sum_f h[f]*W3[f,cols[i],lane])
    for (int i = 0; i < NUM_ELECTRON; ++i) {
      const int idx = sCols[i] * NUM_ELECTRON + lane;  // coalesced over lanes
      float s = b3[idx]
              + h0  * W3[0 * NUM_ORBITAL * NUM_ELECTRON + idx]
              + h1v * W3[1 * NUM_ORBITAL * NUM_ELECTRON + idx]
              + h2v * W3[2 * NUM_ORBITAL * NUM_ELECTRON + idx]
              + h3v * W3[3 * NUM_ORBITAL * NUM_ELECTRON + idx];
      buf[i * LUP + lane] = tanhf(s);
    }

    // Wave-synchronous LU with partial pivoting; lane = column.
    float det = 1.0f;
    for (int k = 0; k < NUM_ELECTRON; ++k) {
      // Lane-parallel pivot search: lane i holds column-k entry of row i
      // (stride LUP=33 -> conflict-free), then 5-step shfl_xor argmax with
      // first-max (lowest index) tie-breaking.
      float cval = buf[lane * LUP + k];
      float cmag = (lane >= k) ? fabsf(cval) : -1.0f;
      int   idx  = lane;
#pragma unroll
      for (int off = 16; off > 0; off >>= 1) {
        const float omag = __shfl_xor(cmag, off);
        const float oval = __shfl_xor(cval, off);
        const int   oidx = __shfl_xor(idx,  off);
        const bool take = (omag > cmag) || (omag == cmag && oidx < idx);
        cmag = take ? omag : cmag;
        cval = take ? oval : cval;
        idx  = take ? oidx : idx;
      }
      const int   p   = idx;   // uniform across lanes
      const float piv = cval;  // uniform across lanes

      if (p != k) {            // row swap (uniform branch)
        const float t = buf[k * LUP + lane];
        buf[k * LUP + lane] = buf[p * LUP + lane];
        buf[p * LUP + lane] = t;
        det = -det;
      }
      det *= piv;
      const float rp = (piv != 0.0f) ? (1.0f / piv) : 0.0f;
      const float pr = buf[k * LUP + lane];
      for (int i = k + 1; i < NUM_ELECTRON; ++i) {
        const float f = buf[i * LUP + k] * rp;            // uniform broadcast
        buf[i * LUP + lane] = fmaf(-f, pr, buf[i * LUP + lane]);
      }
    }
    if (lane == 0) out[base + m] = det;
  }
}

// ---------------------------------------------------------------------------
extern "C" void kernel_launch(void* const* d_in, const int* in_sizes, int n_in,
                              void* d_out, int out_size, void* d_ws, size_t ws_size,
                              hipStream_t stream) {
  (void)n_in; (void)out_size; (void)ws_size;
  const float* x  = (const float*)d_in[0];
  const float* W1 = (const float*)d_in[1];
  const float* b1 = (const float*)d_in[2];
  const float* W2 = (const float*)d_in[3];
  const float* b2 = (const float*)d_in[4];
  const float* W3 = (const float*)d_in[5];
  const float* b3 = (const float*)d_in[6];
  float* out = (float*)d_out;
  int* cols  = (int*)d_ws;   // 32 ints of scratch

  const long total = (long)in_sizes[0];
  const int  batch = in_sizes[0] / NUM_ORBITAL;

  bf_find_cols<<<1, 32, 0, stream>>>(x, cols, total);
  bf_main<<<batch / 16, 256, 0, stream>>>(x, W1, b1, W2, b2, W3, b3, cols, out);
}